// Pow2Quant_67465346285679
// MI455X (gfx1250) — compile-verified
//
#include <hip/hip_runtime.h>
#include <stdint.h>

// Pow2 codebook quantization: out = nearest of {0, ±2^-7 ... ±2^0} to x,
// tie-breaking matching jnp.argmin (first index wins):
//   positive x: ties round toward smaller magnitude (strict >  on mantissa 1.5)
//   negative x: ties round toward larger magnitude  (      >= on mantissa 1.5)
// Closed form via exponent rounding -> ~9 VALU ops/element, memory-bound.
__device__ __forceinline__ float quant_pow2(float x) {
    uint32_t u   = __float_as_uint(x);
    uint32_t sgn = u & 0x80000000u;
    uint32_t a   = u & 0x7FFFFFFFu;   // |x| bits
    uint32_t neg = u >> 31;           // 1 if negative

    // Round |x| to nearest power of two in linear distance:
    // mantissa field M: round up iff M > 0x400000 (pos) / M >= 0x400000 (neg).
    uint32_t rb = (a + 0x003FFFFFu + neg) & 0x7F800000u;
    float r = __uint_as_float(rb);

    // Clamp magnitude into codebook range [2^-7, 1].
    r = fminf(fmaxf(r, 0.0078125f), 1.0f);

    // Zero region around the 0 <-> 2^-7 midpoint (2^-8 = 0x3B800000):
    //   positive: |x| <= 2^-8 -> 0 ; negative: |x| < 2^-8 -> 0.
    uint32_t zthr = 0x3B800000u - neg;

    float out = __uint_as_float(__float_as_uint(r) | sgn);
    return (a <= zthr) ? 0.0f : out;
}

#define TPB 256

// Double-buffered async global->LDS streaming pipeline (CDNA5 ASYNCcnt path).
// Each lane async-copies its own 16B into its private LDS slot, so only
// s_wait_asynccnt ordering is needed (no barriers): async loads complete
// in order within a wave, and the ds_load of a slot fully drains (DScnt)
// before the next overwrite of that slot is even issued.
__global__ __launch_bounds__(TPB)
void Pow2Quant_67465346285679_kernel(const float* __restrict__ in,
                                     float4* __restrict__ out, int n4) {
    __shared__ alignas(16) float4 lbuf[2][TPB];
    const int tid    = (int)threadIdx.x;
    const int stride = (int)(gridDim.x * blockDim.x);
    int i = (int)(blockIdx.x * blockDim.x + tid);

    // 32-bit LDS byte addresses of this lane's two slots
    // (low 32 bits of the generic shared pointer == LDS offset per ISA).
    const uint32_t lds0 = (uint32_t)(uintptr_t)&lbuf[0][tid];
    const uint32_t lds1 = (uint32_t)(uintptr_t)&lbuf[1][tid];

    if (i < n4) {
        uint32_t voff = (uint32_t)i * 16u;   // byte offset, GVS mode
        asm volatile("global_load_async_to_lds_b128 %0, %1, %2"
                     :: "v"(lds0), "v"(voff), "s"(in) : "memory");
    }

    int buf = 0;
    for (; i < n4; i += stride) {
        const int nxt = i + stride;
        if (nxt < n4) {
            uint32_t voff = (uint32_t)nxt * 16u;
            uint32_t ldsn = buf ? lds0 : lds1;
            asm volatile("global_load_async_to_lds_b128 %0, %1, %2"
                         :: "v"(ldsn), "v"(voff), "s"(in) : "memory");
            // two in flight; in-order completion -> oldest is done at <=1
            asm volatile("s_wait_asynccnt 0x1" ::: "memory");
        } else {
            asm volatile("s_wait_asynccnt 0x0" ::: "memory");
        }

        float4 v = lbuf[buf][tid];           // ds_load_b128
        float4 r;
        r.x = quant_pow2(v.x);
        r.y = quant_pow2(v.y);
        r.z = quant_pow2(v.z);
        r.w = quant_pow2(v.w);
        out[i] = r;                          // global_store_b128
        buf ^= 1;
    }
}

__global__ void Pow2Quant_67465346285679_tail(const float* __restrict__ in,
                                              float* __restrict__ out, int n) {
    int i = (int)(blockIdx.x * blockDim.x + threadIdx.x);
    if (i < n) out[i] = quant_pow2(in[i]);
}

extern "C" void kernel_launch(void* const* d_in, const int* in_sizes, int n_in,
                              void* d_out, int out_size, void* d_ws, size_t ws_size,
                              hipStream_t stream) {
    (void)n_in; (void)d_ws; (void)ws_size; (void)out_size;
    const float* x = (const float*)d_in[0];   // 16*256*56*56 f32
    float* out = (float*)d_out;               // f32, same shape
    // d_in[1] (codebook) is a fixed module-level constant; folded into the math.

    const int n   = in_sizes[0];
    const int n4  = n >> 2;
    const int rem = n & 3;

    const int block = TPB;                    // 8 wave32 per block
    long need = ((long)n4 + block - 1) / block;
    int grid = (int)(need < 2048 ? (need > 0 ? need : 1) : 2048);

    if (n4 > 0) {
        Pow2Quant_67465346285679_kernel<<<grid, block, 0, stream>>>(
            x, (float4*)out, n4);
    }
    if (rem > 0) {
        Pow2Quant_67465346285679_tail<<<1, rem, 0, stream>>>(
            x + (size_t)n4 * 4, out + (size_t)n4 * 4, rem);
    }
}